// GINE_77610059039113
// MI455X (gfx1250) — compile-verified
//
#include <hip/hip_runtime.h>

typedef float v2f __attribute__((ext_vector_type(2)));
typedef float v8f __attribute__((ext_vector_type(8)));

#define D 32

// ---------------------------------------------------------------------------
// Kernel 1: h = (1 + eps) * x   (initializes the fp32 accumulator in d_ws)
// ---------------------------------------------------------------------------
__global__ void gine_init_kernel(const float* __restrict__ x,
                                 const float* __restrict__ eps,
                                 float* __restrict__ h, int total) {
  int i = blockIdx.x * blockDim.x + threadIdx.x;
  float scale = 1.0f + eps[0];
  if (i < total) h[i] = scale * x[i];
}

// ---------------------------------------------------------------------------
// Kernel 2: per-(edge,feature) lane: h[dst] += relu(x[src] + e)
// Wave-per-edge: 32 lanes = 32 features, coalesced 128B loads.
// Edge inputs are single-use -> non-temporal load to keep L2 for node tables.
// unsafeAtomicAdd -> hardware global_atomic_add_f32 (resolves in L2).
// ---------------------------------------------------------------------------
__global__ void gine_edge_kernel(const float* __restrict__ x,
                                 const float* __restrict__ e,
                                 const int* __restrict__ src,
                                 const int* __restrict__ dst,
                                 float* __restrict__ h, int nEdges) {
  int tid = blockIdx.x * blockDim.x + threadIdx.x;
  int edge = tid >> 5;
  int f = tid & 31;
  if (edge >= nEdges) return;
  int s = src[edge];
  int d = dst[edge];
  float ev = __builtin_nontemporal_load(&e[(long long)edge * D + f]);
  float m = x[s * D + f] + ev;
  m = fmaxf(m, 0.0f);
  unsafeAtomicAdd(&h[d * D + f], m);
}

// ---------------------------------------------------------------------------
// Kernel 3: out = h @ W^T + b  using V_WMMA_F32_16X16X4_F32.
// One wave handles 16 rows; K=32 -> 8 wmma k-steps of 4; N=32 -> 2 col tiles.
// A 16x4 layout (2 VGPRs): lanes 0-15 hold K={k0,k0+1}, lanes 16-31 K={k0+2,k0+3}.
// B 4x16: lane l<16 holds col n=l for K={k0,k0+1}; lanes 16-31 for K={k0+2,k0+3}.
//   B[k][n] = W^T[k][n] = W[n*32 + k]  (W is [out,in] row-major).
// C/D 16x16 (8 VGPRs): VGPR v holds M=v (lanes 0-15) / M=v+8 (lanes 16-31).
// ---------------------------------------------------------------------------
__global__ void gine_gemm_kernel(const float* __restrict__ h,
                                 const float* __restrict__ W,
                                 const float* __restrict__ bias,
                                 float* __restrict__ out, int nNodes) {
  int wave = (int)((blockIdx.x * blockDim.x + threadIdx.x) >> 5);
  int lane = threadIdx.x & 31;
  int half = lane >> 4;      // 0: lanes 0-15, 1: lanes 16-31
  int l = lane & 15;
  int base = wave * 16;
  if (base >= nNodes) return;

  if (base + 16 <= nNodes) {
    // Full tile: WMMA path (branch is wave-uniform -> EXEC all 1s).
    v2f a[8];
    int row = base + l;
#pragma unroll
    for (int ks = 0; ks < 8; ++ks) {
      a[ks] = *(const v2f*)&h[row * D + ks * 4 + half * 2];
    }
#pragma unroll
    for (int nt = 0; nt < 2; ++nt) {
      int n0 = nt * 16;
      v8f c = {0.f, 0.f, 0.f, 0.f, 0.f, 0.f, 0.f, 0.f};
#pragma unroll
      for (int ks = 0; ks < 8; ++ks) {
        v2f b = *(const v2f*)&W[(n0 + l) * D + ks * 4 + half * 2];
        c = __builtin_amdgcn_wmma_f32_16x16x4_f32(
            /*neg_a=*/false, a[ks], /*neg_b=*/false, b,
            /*c_mod=*/(short)0, c, /*reuse_a=*/false, /*reuse_b=*/false);
      }
      float bv = bias[n0 + l];
#pragma unroll
      for (int v = 0; v < 8; ++v) {
        out[(base + v + half * 8) * D + n0 + l] = c[v] + bv;
      }
    }
  } else {
    // Partial tail tile (not hit for N=100000): scalar per-lane fallback,
    // lane = output column.
    for (int row = base; row < nNodes; ++row) {
      float acc = bias[lane];
      for (int k = 0; k < D; ++k) acc += h[row * D + k] * W[lane * D + k];
      out[row * D + lane] = acc;
    }
  }
}

// ---------------------------------------------------------------------------
// Launch: init -> edge scatter -> WMMA GEMM (stream-ordered dependencies)
// Inputs: 0=node_inputs 1=edge_inputs 2=src 3=dst 4=W 5=b 6=eps
// ---------------------------------------------------------------------------
extern "C" void kernel_launch(void* const* d_in, const int* in_sizes, int n_in,
                              void* d_out, int out_size, void* d_ws, size_t ws_size,
                              hipStream_t stream) {
  const float* x   = (const float*)d_in[0];
  const float* e   = (const float*)d_in[1];
  const int*   src = (const int*)d_in[2];
  const int*   dst = (const int*)d_in[3];
  const float* W   = (const float*)d_in[4];
  const float* b   = (const float*)d_in[5];
  const float* eps = (const float*)d_in[6];
  float* out = (float*)d_out;
  float* h   = (float*)d_ws;   // fp32 accumulator, nNodes*32 floats

  int nNodes = in_sizes[0] / D;
  int nEdges = in_sizes[2];
  int totalH = nNodes * D;

  gine_init_kernel<<<(totalH + 255) / 256, 256, 0, stream>>>(x, eps, h, totalH);

  long long edgeThreads = (long long)nEdges * 32;
  int edgeBlocks = (int)((edgeThreads + 255) / 256);
  gine_edge_kernel<<<edgeBlocks, 256, 0, stream>>>(x, e, src, dst, h, nEdges);

  int tiles = (nNodes + 15) / 16;     // one wave per 16-row tile
  int wavesPerBlock = 8;              // 256 threads = 8 wave32
  int gemmBlocks = (tiles + wavesPerBlock - 1) / wavesPerBlock;
  gine_gemm_kernel<<<gemmBlocks, 256, 0, stream>>>(h, W, b, out, nNodes);
}